// RelGraphConvLayer_81552839016949
// MI455X (gfx1250) — compile-verified
//
#include <hip/hip_runtime.h>

typedef __attribute__((ext_vector_type(2))) float v2f;
typedef __attribute__((ext_vector_type(8))) float v8f;

namespace {
constexpr int kN = 40000;   // nodes
constexpr int kR = 16;      // relations
constexpr int kE = 50000;   // edges per relation
constexpr int kD = 256;     // DIN == DOUT
// GEMM tiling: block tile 64(M) x 256(N), K chunked by 32.
// 8 waves: waveM in {0,1} x waveN in {0..3}; each wave owns 32M x 64N
// (= 2x4 WMMA tiles, 8 v8f accumulators). Each A frag feeds 4 WMMAs,
// each B frag feeds 2 -> few DS ops per WMMA.
constexpr int BM = 64;
constexpr int BN = 256;
constexpr int BK = 32;
constexpr int LDX = 36;           // X tile row stride (floats): conflict-free A b64 reads
constexpr int WROW = 2 * (BN + 16);  // 544 dwords/K-pair row: +16 float2 pad => rows are
                                     // 32 banks apart -> conflict-free B b64 reads
}

// ---------------------------------------------------------------------------
// GEMM: out[m0..m0+64, 0..256] (+)= X @ W
//   USE_BASES: W[k][n] = sum_b coeff[rel,b]*bases[b][k][n], combined in regs
//              while staging (bases = 1 MB, L2-resident).
//   ACCUM:     out += result (self-loop path; out pre-seeded with bias).
// Matrix core: V_WMMA_F32_16X16X4_F32 (full f32 precision).
// LDS W layout is K-pair interleaved: ldsW[p][n] = {W[2p][n], W[2p+1][n]}
// so a B fragment is one aligned ds_load_b64 straight into an even VGPR pair.
// ---------------------------------------------------------------------------
template <bool USE_BASES, bool ACCUM>
__global__ __launch_bounds__(256) void gemm_wmma_f32(
    const float* __restrict__ X,      // [kN, 256]
    const float* __restrict__ Wsrc,   // bases [4,256,256] or loop_weight [256,256]
    const float* __restrict__ coeff,  // [kR, 4] (unused unless USE_BASES)
    int rel,
    float* __restrict__ out)          // [kN, 256]
{
  __shared__ float ldsX[BM * LDX];
  __shared__ float ldsW[(BK / 2) * WROW];

  const int tid   = threadIdx.x;
  const int lane  = tid & 31;
  const int wave  = tid >> 5;      // 8 waves
  const int waveM = wave >> 2;     // 0..1 -> 32-row strip
  const int waveN = wave & 3;      // 0..3 -> 64-col strip
  const int m0    = blockIdx.x * BM;   // 625 blocks; 40000 = 625*64, no tail
  const int laneN = lane & 15;
  const int hi    = lane >> 4;     // ISA A/B frag layout: hi half-wave holds K+2,K+3

  float c0 = 0.f, c1 = 0.f, c2 = 0.f, c3 = 0.f;
  if (USE_BASES) {
    c0 = coeff[rel * 4 + 0];
    c1 = coeff[rel * 4 + 1];
    c2 = coeff[rel * 4 + 2];
    c3 = coeff[rel * 4 + 3];
  }

  v8f acc[2][4] = {};  // [M strip][N tile]

  for (int k0 = 0; k0 < kD; k0 += BK) {
    // ---- stage X tile [BM x BK] ----
    {
      const int row = tid >> 3;       // 0..31
      const int c4  = (tid & 7) * 4;  // 0..28
#pragma unroll
      for (int it = 0; it < 2; ++it) {
        const int m = row + it * 32;
        const float4 v = *(const float4*)&X[(m0 + m) * kD + k0 + c4];
        *(float4*)&ldsX[m * LDX + c4] = v;
        if (k0 + BK < kD)  // prefetch next K-chunk (global_prefetch_b8)
          __builtin_prefetch(&X[(m0 + m) * kD + k0 + BK + c4], 0, 0);
      }
    }
    // ---- stage W tile [BK x BN], K-pair interleaved, basis-combined ----
    {
      const int colB   = (tid & 63) * 4;  // 0..252
      const int kpBase = tid >> 6;        // 0..3
#pragma unroll
      for (int it = 0; it < 4; ++it) {
        const int kp = kpBase + it * 4;   // K pair 0..15
        const int g0 = (k0 + kp * 2) * kD + colB;
        float4 w0, w1;                    // rows 2*kp and 2*kp+1
        if (USE_BASES) {
          const float4 a0 = *(const float4*)&Wsrc[g0];
          const float4 a1 = *(const float4*)&Wsrc[g0 + 1 * kD * kD];
          const float4 a2 = *(const float4*)&Wsrc[g0 + 2 * kD * kD];
          const float4 a3 = *(const float4*)&Wsrc[g0 + 3 * kD * kD];
          const float4 b0v = *(const float4*)&Wsrc[g0 + kD];
          const float4 b1v = *(const float4*)&Wsrc[g0 + kD + 1 * kD * kD];
          const float4 b2v = *(const float4*)&Wsrc[g0 + kD + 2 * kD * kD];
          const float4 b3v = *(const float4*)&Wsrc[g0 + kD + 3 * kD * kD];
          w0.x = c0 * a0.x + c1 * a1.x + c2 * a2.x + c3 * a3.x;
          w0.y = c0 * a0.y + c1 * a1.y + c2 * a2.y + c3 * a3.y;
          w0.z = c0 * a0.z + c1 * a1.z + c2 * a2.z + c3 * a3.z;
          w0.w = c0 * a0.w + c1 * a1.w + c2 * a2.w + c3 * a3.w;
          w1.x = c0 * b0v.x + c1 * b1v.x + c2 * b2v.x + c3 * b3v.x;
          w1.y = c0 * b0v.y + c1 * b1v.y + c2 * b2v.y + c3 * b3v.y;
          w1.z = c0 * b0v.z + c1 * b1v.z + c2 * b2v.z + c3 * b3v.z;
          w1.w = c0 * b0v.w + c1 * b1v.w + c2 * b2v.w + c3 * b3v.w;
        } else {
          w0 = *(const float4*)&Wsrc[g0];
          w1 = *(const float4*)&Wsrc[g0 + kD];
        }
        // interleave: dwords [w0.x,w1.x, w0.y,w1.y, w0.z,w1.z, w0.w,w1.w]
        float* dst = &ldsW[kp * WROW + colB * 2];  // 32B-aligned
        *(float4*)(dst + 0) = make_float4(w0.x, w1.x, w0.y, w1.y);
        *(float4*)(dst + 4) = make_float4(w0.z, w1.z, w0.w, w1.w);
      }
    }
    __syncthreads();

    // ---- WMMA over the K-chunk: 8 k-steps x 8 WMMAs ----
#pragma unroll
    for (int kk = 0; kk < BK; kk += 4) {
      v2f a[2];
#pragma unroll
      for (int ms = 0; ms < 2; ++ms) {
        const int mrow = waveM * 32 + ms * 16 + laneN;
        a[ms] = *(const v2f*)&ldsX[mrow * LDX + kk + hi * 2];  // b64, conflict-free
      }
#pragma unroll
      for (int t = 0; t < 4; ++t) {
        const int n = waveN * 64 + t * 16 + laneN;
        const v2f b = *(const v2f*)&ldsW[((kk >> 1) + hi) * WROW + n * 2];  // b64
        acc[0][t] = __builtin_amdgcn_wmma_f32_16x16x4_f32(
            false, a[0], false, b, (short)0, acc[0][t], false, false);
        acc[1][t] = __builtin_amdgcn_wmma_f32_16x16x4_f32(
            false, a[1], false, b, (short)0, acc[1][t], false, false);
      }
    }
    __syncthreads();
  }

  // ---- writeback per ISA C/D layout: VGPR v -> M = v + (hi ? 8 : 0) ----
#pragma unroll
  for (int ms = 0; ms < 2; ++ms) {
    const int mBase = m0 + waveM * 32 + ms * 16 + hi * 8;
#pragma unroll
    for (int t = 0; t < 4; ++t) {
      const int nCol = waveN * 64 + t * 16 + laneN;
#pragma unroll
      for (int v = 0; v < 8; ++v) {
        const int idx = (mBase + v) * kD + nCol;
        if (ACCUM)
          out[idx] += acc[ms][t][v];
        else
          out[idx] = acc[ms][t][v];
      }
    }
  }
}

// ---------------------------------------------------------------------------
// Edge scatter for one relation: out[dst] += m[src] * rdeg[dst]
// 4 edges per 256-thread block, 64 lanes x float4 per 256-wide row.
// ---------------------------------------------------------------------------
__global__ __launch_bounds__(256) void scatter_edges(
    const float* __restrict__ m,     // [kN, 256]
    const int* __restrict__ esrc,    // [kE] (this relation)
    const int* __restrict__ edst,    // [kE]
    const float* __restrict__ rdeg,  // [kN] (this relation), 1/max(deg,1)
    float* __restrict__ out)         // [kN, 256]
{
  const int e = blockIdx.x * 4 + (threadIdx.x >> 6);
  if (e >= kE) return;
  const int c   = (threadIdx.x & 63) * 4;
  const int src = esrc[e];
  const int dst = edst[e];
  const float s = rdeg[dst];
  const float4 v = *(const float4*)&m[src * kD + c];
  float* o = &out[dst * kD + c];
  atomicAdd(o + 0, v.x * s);
  atomicAdd(o + 1, v.y * s);
  atomicAdd(o + 2, v.z * s);
  atomicAdd(o + 3, v.w * s);
}

__global__ void zero_f32(float* __restrict__ p, int n) {
  const int i = blockIdx.x * blockDim.x + threadIdx.x;
  if (i < n) p[i] = 0.f;
}

__global__ void count_deg(const int* __restrict__ edst,  // [kR*kE]
                          float* __restrict__ deg) {     // [kR*kN]
  const int i = blockIdx.x * blockDim.x + threadIdx.x;
  if (i < kR * kE) {
    const int r = i / kE;
    atomicAdd(&deg[r * kN + edst[i]], 1.0f);
  }
}

__global__ void rcp_clamp(float* __restrict__ deg, int n) {
  const int i = blockIdx.x * blockDim.x + threadIdx.x;
  if (i < n) deg[i] = 1.0f / fmaxf(deg[i], 1.0f);
}

__global__ void init_out_bias(float* __restrict__ out,
                              const float* __restrict__ bias) {
  out[blockIdx.x * kD + threadIdx.x] = bias[threadIdx.x];
}

__global__ void relu_inplace(float* __restrict__ out) {
  const int i = blockIdx.x * kD + threadIdx.x;
  out[i] = fmaxf(out[i], 0.f);
}

// ---------------------------------------------------------------------------
// inputs: 0:x[N,256] 1:bases[4,256,256] 2:coeff[16,4] 3:loop_weight[256,256]
//         4:h_bias[256] 5:edge_src[16,50000] 6:edge_dst[16,50000]
// workspace: deg/rdeg [kR*kN] f32 (2.56 MB) + m buffer [kN*256] f32 (40.96 MB)
// ---------------------------------------------------------------------------
extern "C" void kernel_launch(void* const* d_in, const int* in_sizes, int n_in,
                              void* d_out, int out_size, void* d_ws, size_t ws_size,
                              hipStream_t stream) {
  const float* x     = (const float*)d_in[0];
  const float* bases = (const float*)d_in[1];
  const float* coeff = (const float*)d_in[2];
  const float* loopW = (const float*)d_in[3];
  const float* bias  = (const float*)d_in[4];
  const int*   esrc  = (const int*)d_in[5];
  const int*   edst  = (const int*)d_in[6];
  float* out = (float*)d_out;

  float* deg  = (float*)d_ws;           // kR*kN
  float* mbuf = deg + (size_t)kR * kN;  // kN*kD  (L2-resident handoff buffer)

  // 1) per-relation in-degree -> reciprocal (clamped)
  zero_f32<<<(kR * kN + 255) / 256, 256, 0, stream>>>(deg, kR * kN);
  count_deg<<<(kR * kE + 255) / 256, 256, 0, stream>>>(edst, deg);
  rcp_clamp<<<(kR * kN + 255) / 256, 256, 0, stream>>>(deg, kR * kN);

  // 2) out = bias, then out += x @ loop_weight (f32 WMMA, accumulate mode)
  init_out_bias<<<kN, kD, 0, stream>>>(out, bias);
  dim3 ggrid(kN / BM, 1);
  gemm_wmma_f32<false, true><<<ggrid, 256, 0, stream>>>(x, loopW, nullptr, 0, out);

  // 3) per relation: m = x @ (sum_b coeff[r,b] bases[b]); scatter with 1/deg
  for (int r = 0; r < kR; ++r) {
    gemm_wmma_f32<true, false><<<ggrid, 256, 0, stream>>>(x, bases, coeff, r, mbuf);
    scatter_edges<<<(kE + 3) / 4, 256, 0, stream>>>(
        mbuf, esrc + (size_t)r * kE, edst + (size_t)r * kE, deg + (size_t)r * kN, out);
  }

  // 4) ReLU
  relu_inplace<<<kN, kD, 0, stream>>>(out);
}